// QuantumLayer_41145786695846
// MI455X (gfx1250) — compile-verified
//
#include <hip/hip_runtime.h>
#include <hip/hip_bf16.h>

typedef float v2f __attribute__((ext_vector_type(2)));
typedef float v8f __attribute__((ext_vector_type(8)));

#define NQ    10
#define DIM   1024
#define BATCH 16384
#define EMB   64
#define NL    6
#define NC    32   // batch columns per workgroup (2 N-tiles)

#define WMMA_F32(a, b, c) \
    __builtin_amdgcn_wmma_f32_16x16x4_f32(false, (a), false, (b), (short)0, (c), false, false)

// ---------------------------------------------------------------------------
// Kernel 1: angles = x @ W^T + b ; store cos(angle/2), sin(angle/2)
// ---------------------------------------------------------------------------
__global__ void enc_angles(const float* __restrict__ x, const float* __restrict__ W,
                           const float* __restrict__ bias,
                           float* __restrict__ cs, float* __restrict__ sn) {
    int idx = blockIdx.x * blockDim.x + threadIdx.x;
    if (idx >= BATCH * NQ) return;
    int b = idx / NQ, w = idx % NQ;
    const float* xr = x + b * EMB;
    const float* wr = W + w * EMB;
    float a = bias[w];
#pragma unroll
    for (int e = 0; e < EMB; ++e) a = fmaf(xr[e], wr[e], a);
    float s, c;
    sincosf(0.5f * a, &s, &c);
    cs[idx] = c;
    sn[idx] = s;
}

// ---------------------------------------------------------------------------
// Kernel 2: build the fixed 6-layer unitary, one column (= circuit applied to
// basis state |j>) per workgroup, statevector in LDS. Writes sign-folded
// real planes:  pc(j) even: Arp=Re, Aip=Im ; pc odd: Arp=-Im, Aip=Re.
// Wire w <-> bit (9-w).
// ---------------------------------------------------------------------------
__global__ void build_unitary(const float* __restrict__ weights,
                              float* __restrict__ Arp, float* __restrict__ Aip) {
    __shared__ float re[DIM];
    __shared__ float im[DIM];
    const int j = blockIdx.x;
    const int tid = threadIdx.x;  // 256 threads

    for (int k = tid; k < DIM; k += 256) { re[k] = (k == j) ? 1.f : 0.f; im[k] = 0.f; }
    __syncthreads();

    for (int l = 0; l < NL; ++l) {
        for (int w = 0; w < NQ; ++w) {
            const float* g = weights + (l * NQ + w) * 3;
            float phi = g[0], th = g[1], om = g[2];
            float s, c;   sincosf(0.5f * th, &s, &c);
            float sa, ca; sincosf(0.5f * (phi + om), &sa, &ca);
            float sb, cb; sincosf(0.5f * (phi - om), &sb, &cb);
            float u00r =  ca * c, u00i = -sa * c;
            float u01r = -cb * s, u01i = -sb * s;
            float u10r =  cb * s, u10i = -sb * s;
            float u11r =  ca * c, u11i =  sa * c;
            int tm = 1 << (9 - w);
            for (int p = tid; p < DIM / 2; p += 256) {
                int k0 = ((p & ~(tm - 1)) << 1) | (p & (tm - 1));
                int k1 = k0 | tm;
                float a0r = re[k0], a0i = im[k0], a1r = re[k1], a1i = im[k1];
                re[k0] = u00r * a0r - u00i * a0i + u01r * a1r - u01i * a1i;
                im[k0] = u00r * a0i + u00i * a0r + u01r * a1i + u01i * a1r;
                re[k1] = u10r * a0r - u10i * a0i + u11r * a1r - u11i * a1i;
                im[k1] = u10r * a0i + u10i * a0r + u11r * a1i + u11i * a1r;
            }
            __syncthreads();
        }
        int r = l % (NQ - 1) + 1;
        for (int w = 0; w < NQ; ++w) {
            int cm = 1 << (9 - w);
            int tm = 1 << (9 - ((w + r) % NQ));
            for (int k = tid; k < DIM; k += 256) {
                if ((k & cm) && !(k & tm)) {   // control=1, target=0 : swap pair
                    int k2 = k | tm;
                    float tr = re[k]; re[k] = re[k2]; re[k2] = tr;
                    float ti = im[k]; im[k] = im[k2]; im[k2] = ti;
                }
            }
            __syncthreads();
        }
    }
    const int pc = __popc(j);
    for (int k = tid; k < DIM; k += 256) {
        float rr = re[k], ii = im[k];
        if (pc & 1) { Arp[k * DIM + j] = -ii; Aip[k * DIM + j] = rr; }
        else        { Arp[k * DIM + j] =  rr; Aip[k * DIM + j] = ii; }
    }
}

// ---------------------------------------------------------------------------
// Kernel 3: C_r = Arp @ Btilde, C_i = Aip @ Btilde via V_WMMA_F32_16X16X4_F32.
// Each wave owns a 2x2 block of (M-tile, N-tile) pairs -> 8 WMMAs per K-step
// on 4 A-loads + 2 B-loads, with one-step software pipelining.
// Fused |.|^2 @ Z_SIGNS epilogue, deterministic reduction.
// ---------------------------------------------------------------------------
__global__ __launch_bounds__(128) void wmma_apply(
    const float* __restrict__ Arp, const float* __restrict__ Aip,
    const float* __restrict__ cs,  const float* __restrict__ sn,
    float* __restrict__ out) {
    extern __shared__ float Bt[];           // DIM*NC floats, packed: (j>>1)*64 + col*2 + (j&1)
    __shared__ float scr[128][2 * NQ];
    const int tid = threadIdx.x;
    const int colbase = blockIdx.x * NC;

    // ---- Phase 1: Btilde[j, col] = sign(popcount(j)) * prod_w (bit? s : c)
    {
        int col = tid >> 2;      // 32 columns, 4 threads each
        int sub = tid & 3;
        int b = colbase + col;
        float cv[NQ], sv[NQ];
#pragma unroll
        for (int w = 0; w < NQ; ++w) { cv[w] = cs[b * NQ + w]; sv[w] = sn[b * NQ + w]; }
        for (int jj = 0; jj < DIM / 4; ++jj) {
            int j = sub * (DIM / 4) + jj;
            float m = 1.f;
#pragma unroll
            for (int w = 0; w < NQ; ++w)
                m *= ((j >> (9 - w)) & 1) ? sv[w] : cv[w];
            int p4 = __popc(j) & 3;
            float sgn = (p4 == 1 || p4 == 2) ? -1.f : 1.f;   // (-i)^pc sign folding
            Bt[(j >> 1) * 64 + col * 2 + (j & 1)] = sgn * m;
        }
    }
    __syncthreads();

    // ---- Phase 2: WMMA sweep. A 16x4 layout: lanes 0-15 K=0,1 / lanes 16-31 K=2,3.
    const int wave  = tid >> 5;
    const int lane  = tid & 31;
    const int n     = lane & 15;
    const int khalf = lane >> 4;
    float acc[2 * NQ];
#pragma unroll
    for (int q = 0; q < 2 * NQ; ++q) acc[q] = 0.f;

    for (int mt = wave * 2; mt < DIM / 16; mt += 8) {
        const float* a0 = Arp + (mt * 16 + n) * DIM;
        const float* i0 = Aip + (mt * 16 + n) * DIM;
        const float* a1 = a0 + 16 * DIM;
        const float* i1 = i0 + 16 * DIM;
        if (mt + 8 < DIM / 16) {                 // hint next tile-pair rows toward L2/L0
            __builtin_prefetch(a0 + 128 * DIM, 0, 0);
            __builtin_prefetch(i0 + 128 * DIM, 0, 0);
            __builtin_prefetch(a1 + 128 * DIM, 0, 0);
            __builtin_prefetch(i1 + 128 * DIM, 0, 0);
        }
        v8f c00 = {}, d00 = {}, c01 = {}, d01 = {};
        v8f c10 = {}, d10 = {}, c11 = {}, d11 = {};

        int kk = 2 * khalf;                      // prologue loads (j0 = 0)
        v2f ar0 = *(const v2f*)(a0 + kk), ai0 = *(const v2f*)(i0 + kk);
        v2f ar1 = *(const v2f*)(a1 + kk), ai1 = *(const v2f*)(i1 + kk);
        v2f b0  = *(const v2f*)(&Bt[(kk >> 1) * 64 + n * 2]);
        v2f b1  = *(const v2f*)(&Bt[(kk >> 1) * 64 + (16 + n) * 2]);

        for (int j0 = 0; j0 < DIM; j0 += 4) {
            v2f car0 = ar0, cai0 = ai0, car1 = ar1, cai1 = ai1, cb0 = b0, cb1 = b1;
            if (j0 + 4 < DIM) {                  // software pipeline: fetch next K-step
                int kn = j0 + 4 + 2 * khalf;
                ar0 = *(const v2f*)(a0 + kn); ai0 = *(const v2f*)(i0 + kn);
                ar1 = *(const v2f*)(a1 + kn); ai1 = *(const v2f*)(i1 + kn);
                b0  = *(const v2f*)(&Bt[(kn >> 1) * 64 + n * 2]);
                b1  = *(const v2f*)(&Bt[(kn >> 1) * 64 + (16 + n) * 2]);
            }
            c00 = WMMA_F32(car0, cb0, c00);  d00 = WMMA_F32(cai0, cb0, d00);
            c01 = WMMA_F32(car0, cb1, c01);  d01 = WMMA_F32(cai0, cb1, d01);
            c10 = WMMA_F32(car1, cb0, c10);  d10 = WMMA_F32(cai1, cb0, d10);
            c11 = WMMA_F32(car1, cb1, c11);  d11 = WMMA_F32(cai1, cb1, d11);
        }

        // C/D layout: VGPR r, lane -> M = mt*16 + r + 8*khalf, N = lane&15 (+16 for tile 1)
#pragma unroll
        for (int r = 0; r < 8; ++r) {
            int M0 = mt * 16 + r + 8 * khalf;
            int M1 = M0 + 16;
            float p00 = c00[r] * c00[r] + d00[r] * d00[r];
            float p01 = c01[r] * c01[r] + d01[r] * d01[r];
            float p10 = c10[r] * c10[r] + d10[r] * d10[r];
            float p11 = c11[r] * c11[r] + d11[r] * d11[r];
#pragma unroll
            for (int q = 0; q < NQ; ++q) {
                float s0 = ((M0 >> (9 - q)) & 1) ? -1.f : 1.f;
                float s1 = ((M1 >> (9 - q)) & 1) ? -1.f : 1.f;
                acc[q]      += s0 * p00 + s1 * p10;
                acc[NQ + q] += s0 * p01 + s1 * p11;
            }
        }
    }

    // ---- Phase 3: deterministic fixed-order reduction across the 8 threads/column
#pragma unroll
    for (int q = 0; q < 2 * NQ; ++q) scr[tid][q] = acc[q];
    __syncthreads();
    for (int idx = tid; idx < NC * NQ; idx += 128) {
        int nn = idx / NQ, q = idx % NQ;
        int nt = nn >> 4, nl = nn & 15;
        float s = 0.f;
#pragma unroll
        for (int wv = 0; wv < 4; ++wv)
            s += scr[wv * 32 + nl][nt * NQ + q] + scr[wv * 32 + 16 + nl][nt * NQ + q];
        out[(colbase + nn) * NQ + q] = s;
    }
}

// ---------------------------------------------------------------------------
extern "C" void kernel_launch(void* const* d_in, const int* in_sizes, int n_in,
                              void* d_out, int out_size, void* d_ws, size_t ws_size,
                              hipStream_t stream) {
    const float* x       = (const float*)d_in[0];
    const float* W       = (const float*)d_in[1];
    const float* bias    = (const float*)d_in[2];
    const float* weights = (const float*)d_in[3];
    float* out = (float*)d_out;

    float* wsf = (float*)d_ws;
    float* cs  = wsf;                      // BATCH*NQ
    float* sn  = cs + BATCH * NQ;          // BATCH*NQ
    float* Arp = sn + BATCH * NQ;          // DIM*DIM
    float* Aip = Arp + DIM * DIM;          // DIM*DIM   (total ~9.7 MB)

    enc_angles<<<(BATCH * NQ + 255) / 256, 256, 0, stream>>>(x, W, bias, cs, sn);
    build_unitary<<<DIM, 256, 0, stream>>>(weights, Arp, Aip);
    wmma_apply<<<BATCH / NC, 128, DIM * NC * sizeof(float), stream>>>(Arp, Aip, cs, sn, out);
}